// LatentSliceKernel_67302137528388
// MI455X (gfx1250) — compile-verified
//
#include <hip/hip_runtime.h>

typedef __attribute__((ext_vector_type(4))) float v4f;

namespace {
constexpr int   kD    = 8;
constexpr int   kTMax = 50;      // MAX_RESAMPLINGS
constexpr float kStep = 0.1f;    // STEP_SIZE
}

// One thread == one row (D=8 floats == 32 B == 2x global_load_b128 per array).
// Wave32 covers 1 KB contiguous per array -> fully coalesced HBM streams.
__global__ __launch_bounds__(256, 1) void latent_slice_kernel(
    const float* __restrict__ x,
    const float* __restrict__ s,
    const float* __restrict__ u0,
    const float* __restrict__ u1,
    const float* __restrict__ u2,
    const float* __restrict__ u3,
    const float* __restrict__ uloop,
    float*       __restrict__ out,
    int n)
{
    const int row = blockIdx.x * blockDim.x + threadIdx.x;
    if (row >= n) return;

    const long long base = (long long)row * kD;

    // ---- Phase 1: vectorized b128 loads of the 5 per-row arrays ----
    v4f xv0  = *reinterpret_cast<const v4f*>(x  + base);
    v4f xv1  = *reinterpret_cast<const v4f*>(x  + base + 4);
    v4f sv0  = *reinterpret_cast<const v4f*>(s  + base);
    v4f sv1  = *reinterpret_cast<const v4f*>(s  + base + 4);
    v4f u1v0 = *reinterpret_cast<const v4f*>(u1 + base);
    v4f u1v1 = *reinterpret_cast<const v4f*>(u1 + base + 4);
    v4f u2v0 = *reinterpret_cast<const v4f*>(u2 + base);
    v4f u2v1 = *reinterpret_cast<const v4f*>(u2 + base + 4);
    v4f u3v0 = *reinterpret_cast<const v4f*>(u3 + base);
    v4f u3v1 = *reinterpret_cast<const v4f*>(u3 + base + 4);
    const float u0v = u0[row];

    float xr[kD], sr[kD], u1r[kD], u2r[kD], u3r[kD];
    #pragma unroll
    for (int d = 0; d < 4; ++d) {
        xr[d]  = xv0[d];  xr[d + 4]  = xv1[d];
        sr[d]  = sv0[d];  sr[d + 4]  = sv1[d];
        u1r[d] = u1v0[d]; u1r[d + 4] = u1v1[d];
        u2r[d] = u2v0[d]; u2r[d + 4] = u2v1[d];
        u3r[d] = u3v0[d]; u3r[d + 4] = u3v1[d];
    }

    // y = log1p(-u0) + potential(x);  potential(z) = -0.5 * sum(z*z)
    float potx = 0.0f;
    #pragma unroll
    for (int d = 0; d < kD; ++d) potx += xr[d] * xr[d];
    potx *= -0.5f;
    const float y = log1pf(-u0v) + potx;

    // Initial bracket + first proposal
    float ar[kD], br[kD], xn[kD];
    float potn = 0.0f;
    #pragma unroll
    for (int d = 0; d < kD; ++d) {
        const float l  = u1r[d] * sr[d] + (xr[d] - sr[d] * 0.5f);
        const float sn = 2.0f * fabsf(l - xr[d]) + log1pf(-u2r[d]) / (-kStep);
        ar[d] = l - sn * 0.5f;
        br[d] = l + sn * 0.5f;
        xn[d] = u3r[d] * (br[d] - ar[d]) + ar[d];
        potn += xn[d] * xn[d];
    }
    potn *= -0.5f;
    bool reject = (potn < y);

    // ---- Phase 2: shrink-and-resample loop over the 200 MB u_loop stream ----
    // Once a row accepts, x_new is frozen (reference masks every update with
    // `reject`), so skipping its loads is exactly equivalent. The wave-level
    // ballot exit avoids streaming u_loop[t] for fully-accepted waves at all.
    const float*    up      = uloop + base;
    const long long tstride = (long long)n * kD;   // 1M floats = 4 MB per step

    for (int t = 0; t < kTMax; ++t) {
        if (!__any((int)reject)) break;            // wave-uniform early exit
        if (reject) {
            // Speculative gfx1250 prefetch two steps down the strided stream.
            if (t + 2 < kTMax)
                __builtin_prefetch(up + 2 * tstride, 0, 1);

            // Streaming (non-temporal) b128 loads: read-once, > L2 capacity.
            const v4f* p  = reinterpret_cast<const v4f*>(up);
            const v4f uv0 = __builtin_nontemporal_load(p);
            const v4f uv1 = __builtin_nontemporal_load(p + 1);

            float pn = 0.0f;
            #pragma unroll
            for (int d = 0; d < kD; ++d) {
                const float uv = (d < 4) ? uv0[d] : uv1[d - 4];
                if (xn[d] < xr[d]) ar[d] = xn[d];  // shrink left bound
                else               br[d] = xn[d];  // shrink right bound
                xn[d] = uv * (br[d] - ar[d]) + ar[d];
                pn += xn[d] * xn[d];
            }
            reject = ((-0.5f * pn) < y);
        }
        up += tstride;
    }

    // ---- Store result (b128 x2) ----
    v4f o0, o1;
    #pragma unroll
    for (int d = 0; d < 4; ++d) { o0[d] = xn[d]; o1[d] = xn[d + 4]; }
    *reinterpret_cast<v4f*>(out + base)     = o0;
    *reinterpret_cast<v4f*>(out + base + 4) = o1;
}

extern "C" void kernel_launch(void* const* d_in, const int* in_sizes, int n_in,
                              void* d_out, int out_size, void* d_ws, size_t ws_size,
                              hipStream_t stream) {
    // setup_inputs() order: x, s, u0, u1, u2, u3, u_loop
    const float* x     = (const float*)d_in[0];
    const float* s     = (const float*)d_in[1];
    const float* u0    = (const float*)d_in[2];
    const float* u1    = (const float*)d_in[3];
    const float* u2    = (const float*)d_in[4];
    const float* u3    = (const float*)d_in[5];
    const float* uloop = (const float*)d_in[6];
    float*       out   = (float*)d_out;

    const int n = in_sizes[2];                 // u0 has N elements
    const int threads = 256;                   // 8 wave32 per block
    const int blocks  = (n + threads - 1) / threads;

    latent_slice_kernel<<<blocks, threads, 0, stream>>>(
        x, s, u0, u1, u2, u3, uloop, out, n);
}